// GeomGCN_layer_61220463837361
// MI455X (gfx1250) — compile-verified
//
#include <hip/hip_runtime.h>
#include <hip/hip_bf16.h>

typedef float v2f __attribute__((ext_vector_type(2)));
typedef float v4f __attribute__((ext_vector_type(4)));
typedef float v8f __attribute__((ext_vector_type(8)));

#define D_FEAT  64
#define N_REL   4
#define KDIM    256   // N_REL * D_FEAT
#define OUT_DIM 64

// ---------------------------------------------------------------------------
// Phase 0: zero the h accumulator (float4 stores, one per thread)
// ---------------------------------------------------------------------------
__global__ void geomgcn_zero_h(float* __restrict__ h, long n4) {
    long i = (long)blockIdx.x * blockDim.x + threadIdx.x;
    if (i < n4) {
        ((v4f*)h)[i] = (v4f){0.f, 0.f, 0.f, 0.f};
    }
}

// ---------------------------------------------------------------------------
// Phase 1: edge scatter. One wave32 per edge; each lane handles 2 features.
// h[row, rel*64 + d] += w * x[col, d]   via hardware global_atomic_add_f32.
// ---------------------------------------------------------------------------
__global__ void geomgcn_scatter(const float* __restrict__ x,
                                const int*   __restrict__ ei,    // [2, E]
                                const int*   __restrict__ rel,   // [E]
                                const float* __restrict__ w,     // [E]
                                float*       __restrict__ h,     // [N, 256]
                                int E) {
    long t = (long)blockIdx.x * blockDim.x + threadIdx.x;
    int  e = (int)(t >> 5);
    if (e >= E) return;
    int lane = (int)(t & 31);

    int   row = ei[e];
    int   col = ei[E + e];
    int   r   = rel[e];
    float wt  = w[e];

    v2f xv = *(const v2f*)(x + (long)col * D_FEAT + lane * 2);

    float* dst = h + (long)row * KDIM + r * D_FEAT + lane * 2;
    __hip_atomic_fetch_add(dst,     wt * xv.x, __ATOMIC_RELAXED, __HIP_MEMORY_SCOPE_AGENT);
    __hip_atomic_fetch_add(dst + 1, wt * xv.y, __ATOMIC_RELAXED, __HIP_MEMORY_SCOPE_AGENT);
}

// ---------------------------------------------------------------------------
// Phase 2: out = h @ W^T + b  with V_WMMA_F32_16X16X4_F32 (full fp32 WMMA).
// Block = 8 waves arranged 2(M-tiles) x 4(N-tiles) -> 32 rows x 64 cols.
//
// Per-lane fragment layouts (CDNA5 ISA 7.12.2, 32-bit data):
//   A 16x4 : lanes 0-15 hold M=lane, K={k0,k0+1};  lanes 16-31 K={k0+2,k0+3}
//   B 4x16 : lanes 0-15 hold N=lane, K={k0,k0+1};  lanes 16-31 K={k0+2,k0+3}
//   C 16x16: VGPR v holds M = v + 8*(lane>=16), N = lane&15
// ---------------------------------------------------------------------------
__global__ void geomgcn_gemm_wmma(const float* __restrict__ h,   // [N, 256]
                                  const float* __restrict__ W,   // [64, 256]
                                  const float* __restrict__ b,   // [64]
                                  float*       __restrict__ out, // [N, 64]
                                  int N) {
    int lane   = threadIdx.x & 31;
    int wave   = threadIdx.x >> 5;
    int m_tile = wave >> 2;          // 0..1
    int n_tile = wave & 3;           // 0..3
    int base   = blockIdx.x * 32;

    int l15  = lane & 15;
    int koff = (lane >> 4) << 1;     // 0 for lanes 0-15, 2 for lanes 16-31

    int rowA = base + m_tile * 16 + l15;
    if (rowA >= N) rowA = N - 1;     // clamp loads (EXEC stays all-ones)
    int nB = n_tile * 16 + l15;

    const float* aptr = h + (long)rowA * KDIM + koff;
    const float* bptr = W + (long)nB   * KDIM + koff;

    v8f c = {};
    #pragma unroll 8
    for (int k0 = 0; k0 < KDIM; k0 += 4) {
        v2f a  = *(const v2f*)(aptr + k0);
        v2f bb = *(const v2f*)(bptr + k0);
        // (neg_a, A, neg_b, B, c_mod, C, reuse_a, reuse_b)
        c = __builtin_amdgcn_wmma_f32_16x16x4_f32(
                false, a, false, bb, (short)0, c, false, false);
    }

    float bias  = b[nB];
    int   mhalf = (lane >> 4) * 8;
    int   row0  = base + m_tile * 16 + mhalf;
    float* op   = out + (long)row0 * OUT_DIM + nB;

    if (base + 32 <= N) {
        // Uniform fast path: whole 32-row tile in bounds -> unconditional
        // store clause, no exec-mask churn.
        #pragma unroll
        for (int v = 0; v < 8; ++v) {
            op[(long)v * OUT_DIM] = c[v] + bias;
        }
    } else {
        // Tail block (never taken when N % 32 == 0)
        #pragma unroll
        for (int v = 0; v < 8; ++v) {
            if (row0 + v < N) {
                op[(long)v * OUT_DIM] = c[v] + bias;
            }
        }
    }
}

// ---------------------------------------------------------------------------
extern "C" void kernel_launch(void* const* d_in, const int* in_sizes, int n_in,
                              void* d_out, int out_size, void* d_ws, size_t ws_size,
                              hipStream_t stream) {
    const float* x   = (const float*)d_in[0];   // [N, 64]
    const int*   ei  = (const int*)  d_in[1];   // [2, E]
    const int*   rel = (const int*)  d_in[2];   // [E]
    const float* w   = (const float*)d_in[3];   // [E]
    const float* Wm  = (const float*)d_in[4];   // [64, 256]
    const float* bv  = (const float*)d_in[5];   // [64]

    int N = in_sizes[0] / D_FEAT;
    int E = in_sizes[2];

    float* h   = (float*)d_ws;        // [N, 256] fp32 accumulator (102.4 MB)
    float* out = (float*)d_out;       // [N, 64]

    // Phase 0: zero h
    long n4 = (long)N * KDIM / 4;
    int  zblocks = (int)((n4 + 255) / 256);
    geomgcn_zero_h<<<zblocks, 256, 0, stream>>>(h, n4);

    // Phase 1: scatter-add over edges (one wave per edge)
    long sthreads = (long)E * 32;
    int  sblocks  = (int)((sthreads + 255) / 256);
    geomgcn_scatter<<<sblocks, 256, 0, stream>>>(x, ei, rel, w, h, E);

    // Phase 2: fused GEMM + bias with fp32 WMMA
    int gblocks = (N + 31) / 32;
    geomgcn_gemm_wmma<<<gblocks, 256, 0, stream>>>(h, Wm, bv, out, N);
}